// PointNetFeaturePropagation_61993557950587
// MI455X (gfx1250) — compile-verified
//
#include <hip/hip_runtime.h>
#include <hip/hip_bf16.h>

// ---------------------------------------------------------------------------
// PointNet++ Feature Propagation on gfx1250 (MI455X), wave32 + WMMA fp32.
//
// B=16, N=1024, S=4096, D1=D2=256, C=512, M = B*S = 65536.
// Pipeline:
//   1) knn3          : 3-NN + inverse-distance weights          -> ws.idx3/ws.w3
//   2) build_x       : interp + concat, layout [C, M]           -> d_out (scratch)
//   3) gemm (WMMA)   : Y0 = W0 * X0 + b0                        -> ws.Y
//   4) stats/final   : deterministic per-channel mean/var       -> scale/shift
//   5) bn+relu       : X1 = relu(bn(Y0))                        -> d_out (scratch)
//   6) gemm (WMMA)   : Y1 = W1 * X1 + b1                        -> ws.Y
//   7) stats/final
//   8) bn+relu+perm  : out[b,o,s] = relu(bn(Y1[o, b*S+s]))      -> d_out (final)
//
// GEMM staging uses GLOBAL_LOAD_ASYNC_TO_LDS (ASYNCcnt) with LDS double
// buffering when the async builtins are available; falls back to synchronous
// vector loads otherwise.
// ---------------------------------------------------------------------------

typedef __attribute__((ext_vector_type(2))) float v2f;
typedef __attribute__((ext_vector_type(8))) float v8f;

#define B_DIM 16
#define N_DIM 1024
#define S_DIM 4096
#define D_DIM 256
#define C_DIM 512
#define M_TOT 65536          // B*S
#define K_DIM 512

// GEMM tiling
#define BO 64                // block o-tile
#define BJ 256               // block j-tile
#define KB 32                // k panel depth
#define NPANEL (K_DIM / KB)  // 16
#define WSTR 40              // Wlds row stride (160B rows: 16B aligned for b128)
#define KSTR 38              // Xlds column stride ([j][k] layout, 8B-aligned pairs)

#if defined(__gfx1250__) && \
    __has_builtin(__builtin_amdgcn_global_load_async_to_lds_b128) && \
    __has_builtin(__builtin_amdgcn_global_load_async_to_lds_b32)
#define USE_ASYNC_LDS 1
// builtin param types per hipcc diagnostics:
//   b128 : int __attribute__((__vector_size__(16))) *   (generic pointers)
//   b32  : int *
typedef int v4i_vs __attribute__((__vector_size__(4 * sizeof(int))));
#if __has_builtin(__builtin_amdgcn_s_wait_asynccnt)
#define WAIT_ASYNC() __builtin_amdgcn_s_wait_asynccnt(0)
#else
#define WAIT_ASYNC() asm volatile("s_wait_asynccnt 0" ::: "memory")
#endif
#else
#define USE_ASYNC_LDS 0
#endif

// ---------------------------------------------------------------------------
// Kernel 1: 3-nearest-neighbor + inverse-distance weights.
// One block = 256 query points of one batch; source cloud cached in LDS.
// ---------------------------------------------------------------------------
__global__ __launch_bounds__(256) void knn3_kernel(
    const float* __restrict__ xyz1,   // [B,3,N]
    const float* __restrict__ xyz2,   // [B,3,S]
    int*   __restrict__ idx3,         // [B*S,3]
    float* __restrict__ w3)           // [B*S,3]
{
    __shared__ float sx[N_DIM], sy[N_DIM], sz[N_DIM];
    const int b = blockIdx.y;
    const int s = blockIdx.x * 256 + threadIdx.x;

    const float* base1 = xyz1 + (size_t)b * 3 * N_DIM;
    for (int i = threadIdx.x; i < N_DIM; i += 256) {
        sx[i] = base1[i];
        sy[i] = base1[N_DIM + i];
        sz[i] = base1[2 * N_DIM + i];
    }
    __syncthreads();

    const float* base2 = xyz2 + (size_t)b * 3 * S_DIM;
    const float px = base2[s];
    const float py = base2[S_DIM + s];
    const float pz = base2[2 * S_DIM + s];

    float d0 = 3.4e38f, d1 = 3.4e38f, d2 = 3.4e38f;
    int   i0 = 0, i1 = 0, i2 = 0;
    for (int n = 0; n < N_DIM; ++n) {
        const float dx = px - sx[n];
        const float dy = py - sy[n];
        const float dz = pz - sz[n];
        const float d  = dx * dx + dy * dy + dz * dz;
        if (d < d2) {
            if (d < d1) {
                if (d < d0) { d2 = d1; i2 = i1; d1 = d0; i1 = i0; d0 = d; i0 = n; }
                else        { d2 = d1; i2 = i1; d1 = d;  i1 = n; }
            } else          { d2 = d;  i2 = n; }
        }
    }
    const float r0 = 1.0f / (d0 + 1e-8f);
    const float r1 = 1.0f / (d1 + 1e-8f);
    const float r2 = 1.0f / (d2 + 1e-8f);
    const float rs = 1.0f / (r0 + r1 + r2);

    const size_t j = (size_t)b * S_DIM + s;
    idx3[j * 3 + 0] = i0;  idx3[j * 3 + 1] = i1;  idx3[j * 3 + 2] = i2;
    w3[j * 3 + 0] = r0 * rs; w3[j * 3 + 1] = r1 * rs; w3[j * 3 + 2] = r2 * rs;
}

// ---------------------------------------------------------------------------
// Kernel 2: build X[c, j] (c-major, j = b*S+s):
//   c <  256 : copy of points2[b,c,s]
//   c >= 256 : 3-point weighted interpolation of points1[b, c-256, :]
// ---------------------------------------------------------------------------
__global__ __launch_bounds__(256) void build_x_kernel(
    const float* __restrict__ points1,  // [B,256,N]
    const float* __restrict__ points2,  // [B,256,S]
    const int*   __restrict__ idx3,
    const float* __restrict__ w3,
    float* __restrict__ X)              // [C, M]
{
    const int gid = blockIdx.x * 256 + threadIdx.x;   // < C_DIM * M_TOT
    const int c = gid >> 16;                          // / M_TOT
    const int j = gid & (M_TOT - 1);
    const int b = j >> 12;                            // / S_DIM
    const int s = j & (S_DIM - 1);

    float v;
    if (c < D_DIM) {
        v = points2[((size_t)b * D_DIM + c) * S_DIM + s];
    } else {
        const int d = c - D_DIM;
        const float* row = points1 + ((size_t)b * D_DIM + d) * N_DIM;
        const int jj = j * 3;
        v = w3[jj + 0] * row[idx3[jj + 0]] +
            w3[jj + 1] * row[idx3[jj + 1]] +
            w3[jj + 2] * row[idx3[jj + 2]];
    }
    X[gid] = v;
}

// ---------------------------------------------------------------------------
// Kernel 3: Y[o,j] = sum_c W[o,c]*X[c,j] + bias[o] via V_WMMA_F32_16X16X4_F32.
// Block: 256 threads = 8 waves (2 o-waves x 4 j-waves).
// Block tile 64x256, wave tile 32x64 (2x4 subtiles of 16x16, v8f accums).
// K staged through LDS in double-buffered panels of 32 via async LDS loads.
// LDS layouts chosen so every fragment is ONE aligned ds_load_b64:
//   Wlds[o][k]  stride 40  -> A frag = b64 at (row, kk+2*hi)
//   Xlds[j][k]  stride 38  -> B frag = b64 at (jcol, kk+2*hi)
// Fragment layouts per CDNA5 ISA 7.12.2 (lanes>=16 hold K2/K3).
// ---------------------------------------------------------------------------
__global__ __launch_bounds__(256) void gemm_bias_wmma_kernel(
    const float* __restrict__ W,     // [O, K] row-major
    const float* __restrict__ bias,  // [O]
    const float* __restrict__ X,     // [K, M]
    float* __restrict__ Y)           // [O, M]
{
    __shared__ __attribute__((aligned(16))) float Wlds[2][BO * WSTR];
    __shared__ __attribute__((aligned(16))) float Xlds[2][BJ * KSTR];

    const int tid  = threadIdx.x;
    const int wave = tid >> 5;
    const int lane = tid & 31;
    const int hi   = lane >> 4;
    const int l16  = lane & 15;
    const int wo   = wave & 1;      // 0..1
    const int wj   = wave >> 1;     // 0..3

    const int oBase = (blockIdx.x & 7) * BO;   // 8 o-tiles; consecutive blocks
    const int jBase = (blockIdx.x >> 3) * BJ;  // share a j-tile => L2 reuse of X

    // staging maps (uniform across panels)
    const int wrow = tid >> 2;                 // W: row 0..63
    const int wk0  = (tid & 3) * 8;            // W: k group {0,8,16,24}
    const int xj   = wave * 32 + lane;         // X: column 0..255 (lane-contig j)

    v8f acc[2][4];
    {
        const v8f z = {0.f, 0.f, 0.f, 0.f, 0.f, 0.f, 0.f, 0.f};
#pragma unroll
        for (int a = 0; a < 2; ++a)
#pragma unroll
            for (int b = 0; b < 4; ++b) acc[a][b] = z;
    }

#if USE_ASYNC_LDS
    // -------- async staging: memory -> LDS directly, tracked by ASYNCcnt ----
    auto stage = [&](int p, int bf) {
        const int kBase = p * KB;
        {   // W panel: 2 x b128 per thread
            const float* src = W + (size_t)(oBase + wrow) * K_DIM + kBase + wk0;
            float* dst = &Wlds[bf][wrow * WSTR + wk0];
            __builtin_amdgcn_global_load_async_to_lds_b128(
                (v4i_vs*)src, (v4i_vs*)dst, 0, 0);
            __builtin_amdgcn_global_load_async_to_lds_b128(
                (v4i_vs*)(src + 4), (v4i_vs*)(dst + 4), 0, 0);
        }
        {   // X panel transposed into [j][k]: 32 x b32 per thread; each issue
            // reads 128B contiguous j per wave, scatters k-contig into LDS.
            const float* srcBase = X + (size_t)kBase * M_TOT + jBase + xj;
            float* dstBase = &Xlds[bf][xj * KSTR];
#pragma unroll
            for (int kk = 0; kk < KB; ++kk) {
                __builtin_amdgcn_global_load_async_to_lds_b32(
                    (int*)(srcBase + (size_t)kk * M_TOT),
                    (int*)(dstBase + kk), 0, 0);
            }
        }
    };
#else
    // -------- synchronous staging fallback ---------------------------------
    auto stage = [&](int p, int bf) {
        const int kBase = p * KB;
        {
            const float* src = W + (size_t)(oBase + wrow) * K_DIM + kBase + wk0;
            const float4 wa = *(const float4*)(src);
            const float4 wb = *(const float4*)(src + 4);
            float* dst = &Wlds[bf][wrow * WSTR + wk0];
            dst[0] = wa.x; dst[1] = wa.y; dst[2] = wa.z; dst[3] = wa.w;
            dst[4] = wb.x; dst[5] = wb.y; dst[6] = wb.z; dst[7] = wb.w;
        }
        {
            const float* srcBase = X + (size_t)kBase * M_TOT + jBase + xj;
            float* dstBase = &Xlds[bf][xj * KSTR];
#pragma unroll
            for (int kk = 0; kk < KB; ++kk)
                dstBase[kk] = srcBase[(size_t)kk * M_TOT];
        }
    };
#endif

    auto compute = [&](int bf) {
        const float* Wl = &Wlds[bf][0];
        const float* Xl = &Xlds[bf][0];
        const int rowA0 = (wo * 32 + l16) * WSTR;           // os = 0
        const int rowA1 = (wo * 32 + 16 + l16) * WSTR;      // os = 1
        const int colB  = (wj * 64 + l16) * KSTR;
#pragma unroll
        for (int kk = 0; kk < KB; kk += 4) {
            const int krow = kk + 2 * hi;
            const v2f a0 = *(const v2f*)&Wl[rowA0 + krow];
            const v2f a1 = *(const v2f*)&Wl[rowA1 + krow];
#pragma unroll
            for (int js = 0; js < 4; ++js) {
                const v2f bv = *(const v2f*)&Xl[colB + js * 16 * KSTR + krow];
                acc[0][js] = __builtin_amdgcn_wmma_f32_16x16x4_f32(
                    false, a0, false, bv, (short)0, acc[0][js], false, false);
                acc[1][js] = __builtin_amdgcn_wmma_f32_16x16x4_f32(
                    false, a1, false, bv, (short)0, acc[1][js], false, false);
            }
        }
    };

#if USE_ASYNC_LDS
    stage(0, 0);
    WAIT_ASYNC();
    __syncthreads();
    for (int p = 0; p < NPANEL; ++p) {
        const int bf = p & 1;
        if (p + 1 < NPANEL) stage(p + 1, bf ^ 1);   // prefetch overlaps WMMAs
        compute(bf);
        WAIT_ASYNC();
        __syncthreads();
    }
#else
    for (int p = 0; p < NPANEL; ++p) {
        __syncthreads();
        stage(p, 0);
        __syncthreads();
        compute(0);
    }
#endif

    // --- epilogue: + bias, store [O, M]
#pragma unroll
    for (int os = 0; os < 2; ++os) {
#pragma unroll
        for (int js = 0; js < 4; ++js) {
            const int jcol = jBase + wj * 64 + js * 16 + l16;
#pragma unroll
            for (int r = 0; r < 8; ++r) {
                const int o = oBase + wo * 32 + os * 16 + r + 8 * hi;
                Y[(size_t)o * M_TOT + jcol] = acc[os][js][r] + bias[o];
            }
        }
    }
}

// ---------------------------------------------------------------------------
// Kernel 4: deterministic per-channel sum / sumsq (tree reduce, no fp atomics)
// ---------------------------------------------------------------------------
__global__ __launch_bounds__(256) void stats_kernel(
    const float* __restrict__ Y, float* __restrict__ sum, float* __restrict__ sumsq)
{
    __shared__ float ssum[256], ssq[256];
    const int o = blockIdx.x;
    const float* row = Y + (size_t)o * M_TOT;
    float s = 0.f, q = 0.f;
    for (int j = threadIdx.x * 4; j < M_TOT; j += 256 * 4) {
        const float4 v = *(const float4*)(row + j);
        s += v.x + v.y + v.z + v.w;
        q += v.x * v.x + v.y * v.y + v.z * v.z + v.w * v.w;
    }
    ssum[threadIdx.x] = s; ssq[threadIdx.x] = q;
    __syncthreads();
    for (int st = 128; st > 0; st >>= 1) {
        if (threadIdx.x < st) {
            ssum[threadIdx.x] += ssum[threadIdx.x + st];
            ssq[threadIdx.x]  += ssq[threadIdx.x + st];
        }
        __syncthreads();
    }
    if (threadIdx.x == 0) { sum[o] = ssum[0]; sumsq[o] = ssq[0]; }
}

__global__ void finalize_bn_kernel(
    const float* __restrict__ sum, const float* __restrict__ sumsq,
    const float* __restrict__ g, const float* __restrict__ beta,
    float* __restrict__ scale, float* __restrict__ shift)
{
    const int o = blockIdx.x * blockDim.x + threadIdx.x;
    if (o < C_DIM) {
        const float invM = 1.0f / (float)M_TOT;
        const float m  = sum[o] * invM;
        const float v  = sumsq[o] * invM - m * m;
        const float sc = g[o] * rsqrtf(v + 1e-5f);
        scale[o] = sc;
        shift[o] = beta[o] - m * sc;
    }
}

// ---------------------------------------------------------------------------
// Kernel 5: BN + ReLU, [O,M] -> [O,M] (float4 vectorized)
// ---------------------------------------------------------------------------
__global__ __launch_bounds__(256) void bnrelu_kernel(
    const float* __restrict__ Y, const float* __restrict__ scale,
    const float* __restrict__ shift, float* __restrict__ Xo)
{
    const int gid = blockIdx.x * 256 + threadIdx.x;  // over O*M/4
    const int o   = gid >> 14;                       // M/4 = 16384 vec4 per row
    const float4 y = ((const float4*)Y)[gid];
    const float sc = scale[o], sh = shift[o];
    float4 r;
    r.x = fmaxf(0.f, y.x * sc + sh);
    r.y = fmaxf(0.f, y.y * sc + sh);
    r.z = fmaxf(0.f, y.z * sc + sh);
    r.w = fmaxf(0.f, y.w * sc + sh);
    ((float4*)Xo)[gid] = r;
}

// ---------------------------------------------------------------------------
// Kernel 6: BN + ReLU + permute [O, b*S+s] -> [B, O, S] (final output)
// ---------------------------------------------------------------------------
__global__ __launch_bounds__(256) void bnrelu_final_kernel(
    const float* __restrict__ Y, const float* __restrict__ scale,
    const float* __restrict__ shift, float* __restrict__ out)
{
    const int gid = blockIdx.x * 256 + threadIdx.x;  // over O*M/4
    const int o   = gid >> 14;
    const int rem = gid & 16383;       // float4 index within [B*S]
    const int b   = rem >> 10;         // S/4 = 1024 vec4 per (o,b)
    const int s4  = rem & 1023;

    const float4 y = ((const float4*)Y)[gid];
    const float sc = scale[o], sh = shift[o];
    float4 r;
    r.x = fmaxf(0.f, y.x * sc + sh);
    r.y = fmaxf(0.f, y.y * sc + sh);
    r.z = fmaxf(0.f, y.z * sc + sh);
    r.w = fmaxf(0.f, y.w * sc + sh);

    // out float4 index: (b*C*S + o*S)/4 + s4
    ((float4*)out)[(size_t)b * (C_DIM * S_DIM / 4) + (size_t)o * (S_DIM / 4) + s4] = r;
}

// ---------------------------------------------------------------------------
extern "C" void kernel_launch(void* const* d_in, const int* in_sizes, int n_in,
                              void* d_out, int out_size, void* d_ws, size_t ws_size,
                              hipStream_t stream) {
    (void)in_sizes; (void)n_in; (void)out_size; (void)ws_size;

    const float* xyz1    = (const float*)d_in[0];
    const float* xyz2    = (const float*)d_in[1];
    const float* points1 = (const float*)d_in[2];
    const float* points2 = (const float*)d_in[3];
    const float* w0      = (const float*)d_in[4];
    const float* b0      = (const float*)d_in[5];
    const float* g0      = (const float*)d_in[6];
    const float* beta0   = (const float*)d_in[7];
    const float* w1      = (const float*)d_in[8];
    const float* b1      = (const float*)d_in[9];
    const float* g1      = (const float*)d_in[10];
    const float* beta1   = (const float*)d_in[11];

    float* out = (float*)d_out;            // doubles as X-buffer scratch
    float* ws  = (float*)d_ws;

    float* Y     = ws;                     // [C, M]  = 128 MiB
    float* sum   = ws + (size_t)C_DIM * M_TOT;
    float* sumsq = sum + C_DIM;
    float* scale = sumsq + C_DIM;
    float* shift = scale + C_DIM;
    float* w3    = shift + C_DIM;          // [M,3]
    int*   idx3  = (int*)(w3 + 3 * M_TOT); // [M,3]

    // 1) 3-NN + weights
    knn3_kernel<<<dim3(S_DIM / 256, B_DIM), 256, 0, stream>>>(xyz1, xyz2, idx3, w3);
    // 2) X0 = concat(points2, interp(points1)) in [C, M] -> d_out scratch
    build_x_kernel<<<(C_DIM * M_TOT) / 256, 256, 0, stream>>>(points1, points2, idx3, w3, out);
    // 3) layer 0 GEMM (WMMA fp32)
    gemm_bias_wmma_kernel<<<(C_DIM / BO) * (M_TOT / BJ), 256, 0, stream>>>(w0, b0, out, Y);
    stats_kernel<<<C_DIM, 256, 0, stream>>>(Y, sum, sumsq);
    finalize_bn_kernel<<<2, 256, 0, stream>>>(sum, sumsq, g0, beta0, scale, shift);
    bnrelu_kernel<<<(C_DIM * M_TOT / 4) / 256, 256, 0, stream>>>(Y, scale, shift, out);
    // 4) layer 1 GEMM (WMMA fp32)
    gemm_bias_wmma_kernel<<<(C_DIM / BO) * (M_TOT / BJ), 256, 0, stream>>>(w1, b1, out, Y);
    stats_kernel<<<C_DIM, 256, 0, stream>>>(Y, sum, sumsq);
    finalize_bn_kernel<<<2, 256, 0, stream>>>(sum, sumsq, g1, beta1, scale, shift);
    // 5) BN + ReLU + [O,M]->[B,O,S] permute into final output
    bnrelu_final_kernel<<<(C_DIM * M_TOT / 4) / 256, 256, 0, stream>>>(Y, scale, shift, out);
}